// LstmNNModel_7559142441235
// MI455X (gfx1250) — compile-verified
//
#include <hip/hip_runtime.h>

// LSTM: SEQ=20, BATCH=131072, IN=1, HID=50 -> gates = 200
// fp32 WMMA 16x16x4 for gates = h @ w_hh^T. Each wave owns 32 batch rows
// (two 16-row M-tiles): every B tile (w_hh) loaded from LDS feeds two
// independent WMMA chains, halving ds-per-WMMA and hiding WMMA latency.
// State (h in LDS, c in regs) persistent across steps; no barriers in the
// time loop (all LDS traffic is wave-private; DS ops are in-order per wave).
// Activations use gfx1250's native V_TANH_F32 when available.

#define SEQ   20
#define BATCH 131072
#define HID   50
#define NGATE 200            // 4*HID
#define NT    13             // gate tiles of 16 (padded to 208)
#define NK    13             // K chunks of 4 (padded to 52)
#define HPITCH 52            // h / w_hh LDS pitch (K padded, conflict-free b64)
#define GPITCH 210           // gates LDS pitch (208 + 2: disjoint-bank halves)
#define WAVES  4
#define TPB    (WAVES * 32)
#define ROWS_PER_WAVE  32    // two 16-row M-tiles per wave
#define ROWS_PER_BLOCK (WAVES * ROWS_PER_WAVE)   // 128
#define EPL   50             // elementwise items per lane: 32*50 / 32

typedef float v2f __attribute__((ext_vector_type(2)));
typedef float v8f __attribute__((ext_vector_type(8)));

#if __has_builtin(__builtin_amdgcn_tanhf)
__device__ __forceinline__ float fast_tanh(float x) {
    return __builtin_amdgcn_tanhf(x);                  // v_tanh_f32 (gfx1250)
}
#elif __has_builtin(__builtin_amdgcn_tanh_f32)
__device__ __forceinline__ float fast_tanh(float x) {
    return __builtin_amdgcn_tanh_f32(x);
}
#else
__device__ __forceinline__ float fast_tanh(float x) {
    return fmaf(-2.0f, __builtin_amdgcn_rcpf(__expf(2.0f * x) + 1.0f), 1.0f);
}
#endif

__device__ __forceinline__ float fast_sigmoid(float x) {
#if __has_builtin(__builtin_amdgcn_tanhf) || __has_builtin(__builtin_amdgcn_tanh_f32)
    // sigmoid(x) = 0.5 + 0.5*tanh(x/2): mul + v_tanh + fma
    return fmaf(0.5f, fast_tanh(0.5f * x), 0.5f);
#else
    return __builtin_amdgcn_rcpf(1.0f + __expf(-x));   // v_exp_f32 + v_rcp_f32
#endif
}

__global__ __launch_bounds__(TPB)
void lstm_wmma_kernel(const float* __restrict__ x,
                      const float* __restrict__ w_ih,
                      const float* __restrict__ w_hh,
                      const float* __restrict__ b_ih,
                      const float* __restrict__ b_hh,
                      const float* __restrict__ w_fc,
                      const float* __restrict__ b_fc,
                      float* __restrict__ out) {
    extern __shared__ float smem[];
    float* w_lds = smem;                                   // 208 x 52 (w_hh[n][k], padded)
    float* h_lds = w_lds + 208 * HPITCH;                   // 128 x 52
    float* g_lds = h_lds + ROWS_PER_BLOCK * HPITCH;        // 128 x 210

    const int tid  = threadIdx.x;
    const int lane = tid & 31;
    const int wave = tid >> 5;
    const int lo   = lane & 15;
    const int hi   = lane >> 4;

    // Stage w_hh into LDS as [n][k], zero padded (n: 200->208, k: 50->52).
    for (int idx = tid; idx < 208 * HPITCH; idx += TPB) {
        int n = idx / HPITCH;
        int k = idx - n * HPITCH;
        w_lds[idx] = (n < NGATE && k < HID) ? w_hh[n * HID + k] : 0.0f;
    }
    // h0 = 0 (also zeros the K-padding columns, never rewritten).
    for (int idx = tid; idx < ROWS_PER_BLOCK * HPITCH; idx += TPB) h_lds[idx] = 0.0f;
    __syncthreads();

    float* hw = h_lds + wave * ROWS_PER_WAVE * HPITCH;     // this wave's h: 32 x 52
    float* gw = g_lds + wave * ROWS_PER_WAVE * GPITCH;     // this wave's gates: 32 x 210

    const int base = blockIdx.x * ROWS_PER_BLOCK + wave * ROWS_PER_WAVE;

    // Per-lane, per-N-tile constants: xp[m][n] = x*w_ih[n] + (b_ih+b_hh)[n],
    // n = 16*nt + lo in the C/D layout.
    float wihl[NT], bsl[NT];
#pragma unroll
    for (int nt = 0; nt < NT; ++nt) {
        int n = nt * 16 + lo;
        bool v = (n < NGATE);
        wihl[nt] = v ? w_ih[n] : 0.0f;
        bsl[nt]  = v ? (b_ih[n] + b_hh[n]) : 0.0f;
    }

    // c state: 32 rows x 50 hid = 1600 per wave = 50 per lane.
    float cst[EPL];
#pragma unroll
    for (int i = 0; i < EPL; ++i) cst[i] = 0.0f;

    for (int t = 0; t < SEQ; ++t) {
        // 32 lanes load this wave's 32 x values; broadcast into C-layout rows.
        float xw = x[t * BATCH + base + lane];
        float xr0[8], xr1[8];
#pragma unroll
        for (int j = 0; j < 8; ++j) {
            xr0[j] = __shfl(xw, j + 8 * hi, 32);           // rows 0..15
            xr1[j] = __shfl(xw, 16 + j + 8 * hi, 32);      // rows 16..31
        }

        // Preload A (h) tiles for both M-tiles: lane holds row m, K pair
        // (4*kk + 2*hi, +1). Conflict-free ds_load_b64 (pitch 52).
        v2f a0[NK], a1[NK];
#pragma unroll
        for (int kk = 0; kk < NK; ++kk) {
            a0[kk] = *(const v2f*)(hw + lo * HPITCH + kk * 4 + 2 * hi);
            a1[kk] = *(const v2f*)(hw + (16 + lo) * HPITCH + kk * 4 + 2 * hi);
        }

        // gates[32 x 208] = h @ w_hh^T + xproj; two 16x16 D tiles per nt.
#pragma unroll
        for (int nt = 0; nt < NT; ++nt) {
            v8f acc0, acc1;
#pragma unroll
            for (int j = 0; j < 8; ++j) {
                acc0[j] = fmaf(xr0[j], wihl[nt], bsl[nt]);
                acc1[j] = fmaf(xr1[j], wihl[nt], bsl[nt]);
            }
#pragma unroll
            for (int kk = 0; kk < NK; ++kk) {
                // B tile (4x16): B[k][n] = w_hh[n][k]; one load, two WMMAs.
                v2f b = *(const v2f*)(w_lds + (nt * 16 + lo) * HPITCH + kk * 4 + 2 * hi);
                acc0 = __builtin_amdgcn_wmma_f32_16x16x4_f32(
                    false, a0[kk], false, b, (short)0, acc0, false, false);
                acc1 = __builtin_amdgcn_wmma_f32_16x16x4_f32(
                    false, a1[kk], false, b, (short)0, acc1, false, false);
            }
            // Spill D tiles to wave-private gates LDS (m = j + 8*hi (+16), n = 16*nt + lo).
#pragma unroll
            for (int j = 0; j < 8; ++j) {
                gw[(j + 8 * hi) * GPITCH + nt * 16 + lo]        = acc0[j];
                gw[(16 + j + 8 * hi) * GPITCH + nt * 16 + lo]   = acc1[j];
            }
        }

        // Elementwise LSTM cell update: 50 (m,hid) elements per lane.
#pragma unroll
        for (int i = 0; i < EPL; ++i) {
            int e  = lane + 32 * i;
            int m  = e / HID;
            int hd = e - m * HID;
            const float* gp = gw + m * GPITCH + hd;
            float gi = gp[0];
            float gf = gp[HID];
            float gg = gp[2 * HID];
            float go = gp[3 * HID];
            float is = fast_sigmoid(gi);
            float fs = fast_sigmoid(gf);
            float gt = fast_tanh(gg);
            float os = fast_sigmoid(go);
            float c  = fmaf(fs, cst[i], is * gt);
            cst[i]   = c;
            hw[m * HPITCH + hd] = os * fast_tanh(c);
        }
    }

    // Final FC head: out[b] = h[b] . w_fc + b_fc (all 32 lanes, one row each).
    {
        float acc = b_fc[0];
#pragma unroll
        for (int k = 0; k < HID; ++k)
            acc = fmaf(hw[lane * HPITCH + k], w_fc[k], acc);
        out[base + lane] = acc;
    }
}

extern "C" void kernel_launch(void* const* d_in, const int* in_sizes, int n_in,
                              void* d_out, int out_size, void* d_ws, size_t ws_size,
                              hipStream_t stream) {
    const float* x    = (const float*)d_in[0];
    const float* w_ih = (const float*)d_in[1];
    const float* w_hh = (const float*)d_in[2];
    const float* b_ih = (const float*)d_in[3];
    const float* b_hh = (const float*)d_in[4];
    const float* w_fc = (const float*)d_in[5];
    const float* b_fc = (const float*)d_in[6];
    float* out = (float*)d_out;

    dim3 grid(BATCH / ROWS_PER_BLOCK);   // 1024 blocks
    dim3 block(TPB);                     // 128 threads = 4 waves
    size_t smem = (size_t)(208 * HPITCH + ROWS_PER_BLOCK * HPITCH +
                           ROWS_PER_BLOCK * GPITCH) * sizeof(float); // ~177 KB

    hipLaunchKernelGGL(lstm_wmma_kernel, grid, block, smem, stream,
                       x, w_ih, w_hh, b_ih, b_hh, w_fc, b_fc, out);
}